// HeatmapDecodingLayer_80573586473523
// MI455X (gfx1250) — compile-verified
//
#include <hip/hip_runtime.h>
#include <hip/hip_bf16.h>

typedef __attribute__((ext_vector_type(2))) float v2f;
typedef __attribute__((ext_vector_type(8))) float v8f;

#define BB 32
#define HH 256
#define WW 256
#define CC 16
#define HW (HH * WW)              // 65536
#define SLICES 16
#define SLICE_POS (HW / SLICES)   // 4096

// workspace layout (float offsets)
#define WS_SLICEMAX 0                                   // BB*SLICES*CC = 8192
#define WS_MAX      (WS_SLICEMAX + BB * SLICES * CC)    // 512
#define WS_S        (WS_MAX + BB * CC)                  // 512
#define WS_SX       (WS_S + BB * CC)                    // 512
#define WS_SY       (WS_SX + BB * CC)                   // 512

__device__ __forceinline__ float4 fmax4(float4 a, float4 b) {
    return make_float4(fmaxf(a.x, b.x), fmaxf(a.y, b.y),
                       fmaxf(a.z, b.z), fmaxf(a.w, b.w));
}

// Pass 1: per-(b,c) max over a spatial slice. One block per (b, slice).
__global__ __launch_bounds__(256) void hm_max(const float* __restrict__ x,
                                              float* __restrict__ ws) {
    const int blk = blockIdx.x;
    const int b = blk / SLICES, slice = blk % SLICES;
    const int t = threadIdx.x;
    const int cg = t & 3;      // channel group: channels 4*cg .. 4*cg+3
    const int lp = t >> 2;     // 0..63 position lane within iteration

    const float4* xb = reinterpret_cast<const float4*>(x) + (size_t)b * HW * 4;
    float4 m = make_float4(-3.4e38f, -3.4e38f, -3.4e38f, -3.4e38f);
    const int base = slice * SLICE_POS;
    #pragma unroll 4
    for (int it = 0; it < SLICE_POS / 64; ++it) {
        const int pos = base + it * 64 + lp;
        m = fmax4(m, xb[(size_t)pos * 4 + cg]);
    }

    __shared__ float4 red[256];
    red[t] = m;
    __syncthreads();
    for (int s = 128; s >= 4; s >>= 1) {
        if (t < s) red[t] = fmax4(red[t], red[t + s]);
        __syncthreads();
    }
    if (t < 4) {
        const float4 m4 = red[t];
        float* o = ws + WS_SLICEMAX + ((size_t)(b * SLICES + slice)) * CC + t * 4;
        o[0] = m4.x; o[1] = m4.y; o[2] = m4.z; o[3] = m4.w;
    }
}

// Pass 1b: fold slice maxima -> final per-(b,c) max; zero accumulators.
__global__ __launch_bounds__(256) void hm_maxreduce(float* __restrict__ ws) {
    const int id = blockIdx.x * blockDim.x + threadIdx.x;
    if (id >= BB * CC) return;
    const int b = id >> 4, c = id & 15;
    float m = -3.4e38f;
    #pragma unroll
    for (int s = 0; s < SLICES; ++s)
        m = fmaxf(m, ws[WS_SLICEMAX + (b * SLICES + s) * CC + c]);
    ws[WS_MAX + id] = m;
    ws[WS_S + id] = 0.0f;
    ws[WS_SX + id] = 0.0f;
    ws[WS_SY + id] = 0.0f;
}

// Pass 2: WMMA-accumulated weighted sums (V_WMMA_F32_16X16X4_F32).
//   A (16x4 f32):  A[c][k] = exp2(K2*x[pos+k, c] - K2*max_c)
//                  (lanes 0-15 hold K=0,1; lanes 16-31 hold K=2,3)
//   B (4x16 f32):  B[k][n] = n==0 ? 1 : n==1 ? w(pos+k) : n==2 ? h(pos+k) : 0
//                  (unscaled integer coords; /255 applied in finalize)
//   D (16x16 f32): col 0 = S, col 1 = Sw, col 2 = Sh per channel row.
// A 4-position group is 4-aligned and rows are 256 wide, so h is constant
// within a group and w(pos+k) = (p&255) + k  ->  B is 4 FMAs from per-lane
// loop-invariant masks.
__global__ __launch_bounds__(256) void hm_moments(const float* __restrict__ x,
                                                  float* __restrict__ ws) {
    const int blk = blockIdx.x;
    const int b = blk / SLICES, slice = blk % SLICES;
    const int t = threadIdx.x;
    const int wave = t >> 5;
    const int lane = t & 31;
    const int n = lane & 15;      // channel (A row / B column)
    const int half = lane >> 4;   // 0 -> K=0,1 ; 1 -> K=2,3
    const int k0 = half * 2;

    const float K2 = 512.0f * 1.44269504088896340736f;  // 512 * log2(e)
    const float negmk = -K2 * ws[WS_MAX + b * CC + n];

    // per-lane loop-invariant B-construction masks
    const float m1 = (n == 1) ? 1.0f : 0.0f;            // x-coord column
    const float m2 = (n == 2) ? 1.0f : 0.0f;            // y-coord column
    const float base0 = ((n == 0) ? 1.0f : 0.0f) + m1 * (float)k0;
    const float base1 = ((n == 0) ? 1.0f : 0.0f) + m1 * (float)(k0 + 1);

    const float* xb = x + (size_t)b * HW * CC;
    v8f acc = {0.f, 0.f, 0.f, 0.f, 0.f, 0.f, 0.f, 0.f};

    const int pbase = slice * SLICE_POS;
    for (int it = 0; it < SLICE_POS / 32; ++it) {   // 8 waves * 4 positions
        const int p = pbase + it * 32 + wave * 4;   // group base position
        const int pos_a = p + k0;                   // this lane's first K pos
        const size_t ia = (size_t)pos_a * CC + n;
        const float fa = xb[ia];
        const float fb = xb[ia + CC];

        v2f A;
        A.x = __builtin_amdgcn_exp2f(__builtin_fmaf(fa, K2, negmk));
        A.y = __builtin_amdgcn_exp2f(__builtin_fmaf(fb, K2, negmk));

        const float wbf = (float)(p & (WW - 1));    // group base column
        const float hf  = (float)(p >> 8);          // group row (constant in group)
        v2f Bv;
        Bv.x = __builtin_fmaf(m1, wbf, __builtin_fmaf(m2, hf, base0));
        Bv.y = __builtin_fmaf(m1, wbf, __builtin_fmaf(m2, hf, base1));

        acc = __builtin_amdgcn_wmma_f32_16x16x4_f32(
            /*neg_a=*/false, A, /*neg_b=*/false, Bv,
            /*c_mod=*/(short)0, acc, /*reuse_a=*/false, /*reuse_b=*/false);
    }

    // D layout: VGPR r -> row r (lanes 0-15) / row r+8 (lanes 16-31), col = n.
    if (n <= 2) {
        float* arr = ws + (n == 0 ? WS_S : (n == 1 ? WS_SX : WS_SY)) + b * CC + half * 8;
        #pragma unroll
        for (int r = 0; r < 8; ++r) atomicAdd(&arr[r], acc[r]);
    }
}

// Pass 3: finalize — centroid coords (apply 1/255 scale) + bilinear value read.
__global__ __launch_bounds__(256) void hm_finalize(const float* __restrict__ x,
                                                   const float* __restrict__ ws,
                                                   float* __restrict__ out) {
    const int id = blockIdx.x * blockDim.x + threadIdx.x;
    if (id >= BB * CC) return;
    const int b = id >> 4, c = id & 15;

    const float S  = ws[WS_S + id];
    const float Sw = ws[WS_SX + id];
    const float Sh = ws[WS_SY + id];
    const float inv = 1.0f / S;
    const float px = Sw * inv;                  // in [0, 255] (pixel units)
    const float py = Sh * inv;
    const float cx = px * (1.0f / (float)(WW - 1));
    const float cy = py * (1.0f / (float)(HH - 1));

    int x0 = (int)floorf(px); x0 = min(max(x0, 0), WW - 1);
    int y0 = (int)floorf(py); y0 = min(max(y0, 0), HH - 1);
    const int x1 = min(x0 + 1, WW - 1);
    const int y1 = min(y0 + 1, HH - 1);
    const float fx = px - (float)x0;
    const float fy = py - (float)y0;

    const float* xb = x + (size_t)b * HW * CC + c;
    const float g00 = xb[((size_t)y0 * WW + x0) * CC];
    const float g01 = xb[((size_t)y0 * WW + x1) * CC];
    const float g10 = xb[((size_t)y1 * WW + x0) * CC];
    const float g11 = xb[((size_t)y1 * WW + x1) * CC];
    const float val = g00 * (1.0f - fx) * (1.0f - fy)
                    + g01 * fx * (1.0f - fy)
                    + g10 * (1.0f - fx) * fy
                    + g11 * fx * fy;

    out[id * 3 + 0] = cx;
    out[id * 3 + 1] = cy;
    out[id * 3 + 2] = val;
}

extern "C" void kernel_launch(void* const* d_in, const int* in_sizes, int n_in,
                              void* d_out, int out_size, void* d_ws, size_t ws_size,
                              hipStream_t stream) {
    (void)in_sizes; (void)n_in; (void)out_size; (void)ws_size;
    const float* x = (const float*)d_in[0];
    float* out = (float*)d_out;
    float* ws = (float*)d_ws;

    hm_max<<<BB * SLICES, 256, 0, stream>>>(x, ws);
    hm_maxreduce<<<2, 256, 0, stream>>>(ws);
    hm_moments<<<BB * SLICES, 256, 0, stream>>>(x, ws);
    hm_finalize<<<2, 256, 0, stream>>>(x, ws, out);
}